// CGConv_64707977282171
// MI455X (gfx1250) — compile-verified
//
#include <hip/hip_runtime.h>
#include <math.h>

typedef float v2f __attribute__((ext_vector_type(2)));
typedef float v8f __attribute__((ext_vector_type(8)));

#define N_NODES  25000
#define N_EDGES  400000
#define CH       128
#define ECH      64
#define KDIM     320          // 2*CH + ECH
#define MBLK     64           // edges per workgroup (400000 / 64 = 6250 exact)
#define ZSTRIDE  324          // padded LDS row stride (floats): 324%64==4 -> conflict-free
#define NTHREADS 256          // 8 waves (wave32)

// ---------------------------------------------------------------- residual init
__global__ void cgconv_copy_kernel(const float* __restrict__ x,
                                   float* __restrict__ out, int n4) {
  int i = blockIdx.x * blockDim.x + threadIdx.x;
  if (i < n4) ((float4*)out)[i] = ((const float4*)x)[i];
}

// ---------------------------------------------------------------- fused edge GEMM
__global__ __launch_bounds__(NTHREADS, 1)
void cgconv_edge_kernel(const float* __restrict__ x,
                        const long long* __restrict__ ei,   // [2, E] int64
                        const float* __restrict__ ea,       // [E, ECH]
                        const float* __restrict__ gw,       // [CH, KDIM] row-major
                        const float* __restrict__ gb,       // [CH]
                        const float* __restrict__ mw,       // [CH, KDIM]
                        const float* __restrict__ mb,       // [CH]
                        float* __restrict__ out) {          // [N, CH]
  extern __shared__ float smem[];
  float* zs     = smem;                             // MBLK x ZSTRIDE
  int*   cols_s = (int*)(smem + MBLK * ZSTRIDE);    // MBLK
  int*   rows_s = cols_s + MBLK;                    // MBLK

  const int tid   = threadIdx.x;
  const int lane  = tid & 31;
  const int wv    = tid >> 5;            // 0..7
  const int ebase = blockIdx.x * MBLK;

  // stage edge endpoints (int64 -> int32)
  if (tid < MBLK) {
    rows_s[tid] = (int)ei[ebase + tid];
    cols_s[tid] = (int)ei[N_EDGES + ebase + tid];
  }
  // warm the weight matrices into cache (L2-resident, 320 KB total)
  if (tid < CH) {
    __builtin_prefetch(&gw[(size_t)tid * KDIM], 0, 1);
    __builtin_prefetch(&mw[(size_t)tid * KDIM], 0, 1);
  }
  __syncthreads();

  // gather Z = [x[row] | x[col] | edge_attr] into LDS, float4 granularity
  // KDIM/4 = 80 float4 slots per edge row
  for (int q = tid; q < MBLK * (KDIM / 4); q += NTHREADS) {
    int e  = q / (KDIM / 4);
    int qq = q - e * (KDIM / 4);        // 0..79
    float4 v;
    if (qq < CH / 4)                    // 0..31  -> x[row]
      v = ((const float4*)(x + (size_t)rows_s[e] * CH))[qq];
    else if (qq < CH / 2)               // 32..63 -> x[col]
      v = ((const float4*)(x + (size_t)cols_s[e] * CH))[qq - CH / 4];
    else                                // 64..79 -> edge_attr
      v = ((const float4*)(ea + (size_t)(ebase + e) * ECH))[qq - CH / 2];
    ((float4*)(zs + e * ZSTRIDE))[qq] = v;
  }
  __syncthreads();

  // ---- WMMA fragment coordinates (f32 16x16x4, wave32) ----
  const int nloc = lane & 15;           // N (B/C/D) and M (A) lane coordinate
  const int koff = (lane >> 4) << 1;    // lanes 0-15: K{0,1}; lanes 16-31: K{2,3}
  const int c    = wv * 16 + nloc;      // output column 0..127 (gate AND msg)

  const v8f vzero = {0.f, 0.f, 0.f, 0.f, 0.f, 0.f, 0.f, 0.f};
  v8f accG[4], accM[4];
#pragma unroll
  for (int r = 0; r < 4; ++r) { accG[r] = vzero; accM[r] = vzero; }

  // B fragments come straight from global: W[c][k..k+1] is contiguous (b64)
  const float2* gwp = (const float2*)(gw + (size_t)c * KDIM);
  const float2* mwp = (const float2*)(mw + (size_t)c * KDIM);

#pragma unroll 2
  for (int k = 0; k < KDIM; k += 4) {
    const int kk = (k + koff) >> 1;
    float2 tg = gwp[kk];
    float2 tm = mwp[kk];
    v2f bgv; bgv.x = tg.x; bgv.y = tg.y;
    v2f bmv; bmv.x = tm.x; bmv.y = tm.y;
#pragma unroll
    for (int r = 0; r < 4; ++r) {
      const float2 ta = *(const float2*)(zs + (r * 16 + nloc) * ZSTRIDE + k + koff);
      v2f av; av.x = ta.x; av.y = ta.y;
      accG[r] = __builtin_amdgcn_wmma_f32_16x16x4_f32(
          false, av, false, bgv, (short)0, accG[r], false, false);
      accM[r] = __builtin_amdgcn_wmma_f32_16x16x4_f32(
          false, av, false, bmv, (short)0, accM[r], false, false);
    }
  }

  // ---- epilogue: bias, sigmoid*softplus, scatter-add by col ----
  // Use hardware approximation ops (v_rcp_f32 / v_log_f32 / v_exp_f32) instead
  // of the IEEE divide expansion + log1p libcall: ~1ulp, far below GEMM noise.
  const float gbias  = gb[c];
  const float mbias  = mb[c];
  const int   mshift = (lane >> 4) << 3;   // C/D: lanes>=16 hold M = j+8

#pragma unroll
  for (int r = 0; r < 4; ++r) {
#pragma unroll
    for (int j = 0; j < 8; ++j) {
      const int m = j + mshift;
      float g = accG[r][j] + gbias;
      float t = accM[r][j] + mbias;
      float gate = __builtin_amdgcn_rcpf(1.0f + __expf(-g));        // sigmoid
      float sp   = fmaxf(t, 0.0f) + __logf(1.0f + __expf(-fabsf(t))); // softplus
      float val  = gate * sp;
      int node = cols_s[r * 16 + m];
      unsafeAtomicAdd(out + (size_t)node * CH + c, val);  // global_atomic_add_f32
    }
  }
}

// ---------------------------------------------------------------- launcher
extern "C" void kernel_launch(void* const* d_in, const int* in_sizes, int n_in,
                              void* d_out, int out_size, void* d_ws, size_t ws_size,
                              hipStream_t stream) {
  const float*     x  = (const float*)d_in[0];
  const long long* ei = (const long long*)d_in[1];
  const float*     ea = (const float*)d_in[2];
  const float*     gw = (const float*)d_in[3];
  const float*     gb = (const float*)d_in[4];
  const float*     mw = (const float*)d_in[5];
  const float*     mb = (const float*)d_in[6];
  float* out = (float*)d_out;

  const int n4 = (N_NODES * CH) / 4;
  cgconv_copy_kernel<<<(n4 + NTHREADS - 1) / NTHREADS, NTHREADS, 0, stream>>>(x, out, n4);

  const size_t shmem = (size_t)(MBLK * ZSTRIDE) * sizeof(float) + 2 * MBLK * sizeof(int);
  cgconv_edge_kernel<<<N_EDGES / MBLK, NTHREADS, shmem, stream>>>(
      x, ei, ea, gw, gb, mw, mb, out);
}